// SymmetricContraction_72481868087673
// MI455X (gfx1250) — compile-verified
//
#include <hip/hip_runtime.h>

typedef float v2f __attribute__((ext_vector_type(2)));
typedef float v4f __attribute__((ext_vector_type(4)));
typedef float v8f __attribute__((ext_vector_type(8)));

#define D_IN   9
#define D_OUT  4
#define MUL    128
#define NB     1024
#define P1     3
#define P2     11
#define P3     23
#define NPAR   37     // P1+P2+P3
#define NMONO  219    // 9 + 45 + 165
#define KP     220    // NMONO padded to multiple of 4
#define KBLK   55     // KP/4
#define NCOL   148    // 4*(3+11+23)
#define NP     160    // NCOL padded to 10 tiles of 16
#define WAVES  8
#define ROWS   16

#if defined(__has_builtin)
#if __has_builtin(__builtin_amdgcn_global_load_async_to_lds_b128)
#define USE_ASYNC_LDS 1
#endif
#endif

#ifdef USE_ASYNC_LDS
// Types per hipcc diagnostic: param0 is 'int __attribute__((vector_size(16))) __device__ *'
typedef int async_b128 __attribute__((vector_size(16)));
typedef __attribute__((address_space(1))) async_b128* async_gptr;
typedef __attribute__((address_space(3))) async_b128* async_lptr;
#endif

// ---------------------------------------------------------------------------
// Prep: fold U1/U2/U3 (with symmetry multiplicities) into a packed coefficient
// matrix laid out for WMMA B-fragments: packed[(kb*NP + n)*4 + kk] = C[4kb+kk][n]
// so each lane's (K, K+1) pair for a fragment is a contiguous 8-byte LDS load.
// ---------------------------------------------------------------------------
__global__ void build_cmat_kernel(const float* __restrict__ U1,
                                  const float* __restrict__ U2,
                                  const float* __restrict__ U3,
                                  float* __restrict__ packed)
{
    int idx = blockIdx.x * blockDim.x + threadIdx.x;
    if (idx >= KP * NP) return;
    int kk   = idx & 3;
    int rest = idx >> 2;
    int n    = rest % NP;
    int kb   = rest / NP;
    int K    = kb * 4 + kk;

    float v = 0.0f;
    if (K < NMONO && n < NCOL) {
        if (K < 9) {                       // order-1 monomial x_i
            if (n < 12) {
                int i = K, o = n / P1, p = n % P1;
                v = U1[(i * D_OUT + o) * P1 + p];
            }
        } else if (K < 54) {               // order-2 monomial x_i x_j, i<=j
            if (n >= 12 && n < 56) {
                int q = K - 9, i = 0;
                while (q >= 9 - i) { q -= 9 - i; ++i; }
                int j = i + q;
                int c = n - 12, o = c / P2, p = c % P2;
                float m = (i == j) ? 1.0f : 2.0f;
                v = m * U2[((i * D_IN + j) * D_OUT + o) * P2 + p];
            }
        } else {                           // order-3 monomial x_i x_j x_l, i<=j<=l
            if (n >= 56) {
                int q = K - 54;
                int i = 0, j = 0, l = 0;
                bool done = false;
                for (int ii = 0; ii < 9 && !done; ++ii) {
                    for (int jj = ii; jj < 9 && !done; ++jj) {
                        int cnt = 9 - jj;
                        if (q < cnt) { i = ii; j = jj; l = jj + q; done = true; }
                        else q -= cnt;
                    }
                }
                float m = (i == j && j == l) ? 1.0f : ((i == j || j == l) ? 3.0f : 6.0f);
                int c = n - 56, o = c / P3, p = c % P3;
                v = m * U3[(((i * D_IN + j) * D_IN + l) * D_OUT + o) * P3 + p];
            }
        }
    }
    packed[(kb * NP + n) * 4 + kk] = v;
}

// ---------------------------------------------------------------------------
// Main fused kernel: monomials -> fp32 WMMA GEMM (219 x 148) -> p-contraction.
// 256 threads = 8 waves; each wave owns 16 consecutive (b,k) rows.
// ---------------------------------------------------------------------------
__global__ __launch_bounds__(256) void sym_contract_kernel(
    const float* __restrict__ x,
    const int*   __restrict__ indices,
    const float* __restrict__ weight,
    const float* __restrict__ packed,
    float*       __restrict__ out)
{
    extern __shared__ float smem[];
    float* sC = smem;                         // KP*NP packed coefficients
    float* sA = smem + KP * NP;               // WAVES * ROWS * KP monomials
    float* sW = sA + WAVES * ROWS * KP;       // WAVES * NPAR * 16 weights

    const int tid  = threadIdx.x;
    const int wave = tid >> 5;
    const int lane = tid & 31;
    const int lh   = lane >> 4;               // lane half (0/1)
    const int rowA = lane & 15;               // row within wave tile

    // Hint the table into cache (gfx1250 global_prefetch_b8).
    __builtin_prefetch(packed + tid * 64, 0, 1);

    // ---- coefficient matrix -> LDS ----
#ifdef USE_ASYNC_LDS
    // CDNA5 async global->LDS (GLOBAL_LOAD_ASYNC_TO_LDS_B128, ASYNCcnt):
    // bypasses VGPRs; overlaps with monomial generation below.
    for (int i4 = tid; i4 < (KP * NP) / 4; i4 += 256) {
        async_gptr gsrc = (async_gptr)(uintptr_t)(packed + 4 * i4);
        async_lptr ldst = (async_lptr)(unsigned int)(uintptr_t)(sC + 4 * i4);
        __builtin_amdgcn_global_load_async_to_lds_b128(gsrc, ldst, 0, 0);
    }
#else
    for (int i4 = tid; i4 < (KP * NP) / 4; i4 += 256)
        ((v4f*)sC)[i4] = ((const v4f*)packed)[i4];
#endif

    const int rowBase = (blockIdx.x * WAVES + wave) * ROWS;   // 16 | 128 => one b per tile
    const int b     = rowBase / MUL;
    const int kBase = rowBase % MUL;
    const int e     = indices[b];

    // Stage per-tile weights: weight[e, p, kBase + 0..15]
    float* myW = sW + wave * NPAR * 16;
    for (int i = lane; i < NPAR * 16; i += 32) {
        int p = i >> 4, kk = i & 15;
        myW[i] = weight[(e * NPAR + p) * MUL + kBase + kk];
    }

    // ---- monomial generation into LDS (A tile, 16 x KP row-major) ----
    float* myA  = sA + wave * ROWS * KP;
    float* arow = myA + rowA * KP;
    float xv[9];
    const float* xb = x + b * (D_IN * MUL) + kBase + rowA;    // coalesced over lanes
    #pragma unroll
    for (int i = 0; i < 9; ++i) xv[i] = xb[i * MUL];

    if (lh == 0) {                 // orders 1 and 2  (54 values) + pad
        int idx = 0;
        #pragma unroll
        for (int i = 0; i < 9; ++i) arow[idx++] = xv[i];
        #pragma unroll
        for (int i = 0; i < 9; ++i)
            #pragma unroll
            for (int j = i; j < 9; ++j) arow[idx++] = xv[i] * xv[j];
        arow[KP - 1] = 0.0f;       // zero the K=219 pad (avoid NaN*0)
    } else {                       // order 3 (165 values)
        int idx = 54;
        #pragma unroll
        for (int i = 0; i < 9; ++i)
            #pragma unroll
            for (int j = i; j < 9; ++j) {
                float pij = xv[i] * xv[j];
                #pragma unroll
                for (int l = j; l < 9; ++l) arow[idx++] = pij * xv[l];
            }
    }

#ifdef USE_ASYNC_LDS
    // Drain this wave's async copies before the workgroup barrier.
#if __has_builtin(__builtin_amdgcn_s_wait_asynccnt)
    __builtin_amdgcn_s_wait_asynccnt(0);
#else
    asm volatile("s_wait_asynccnt 0x0" ::: "memory");
#endif
#endif
    __syncthreads();

    // ---- fp32 WMMA GEMM: [16 x KP] x [KP x NP] -> [16 x NP] ----
    v8f acc[10] = {};
    const float* aBase = myA + rowA * KP + 2 * lh;   // A 16x4 frag: V0=K0/K2, V1=K1/K3
    for (int kb = 0; kb < KBLK; ++kb) {
        v2f afr = *(const v2f*)(aBase + kb * 4);
        #pragma unroll
        for (int t = 0; t < 10; ++t) {
            v2f bfr = *(const v2f*)(sC + (kb * NP + t * 16 + rowA) * 4 + 2 * lh);
            acc[t] = __builtin_amdgcn_wmma_f32_16x16x4_f32(
                false, afr, false, bfr, (short)0, acc[t], false, false);
        }
    }
    __syncthreads();

    // ---- dump C tiles to LDS (reuse A region): T[16][NP] ----
    float* myT = myA;
    #pragma unroll
    for (int t = 0; t < 10; ++t)
        #pragma unroll
        for (int vg = 0; vg < 8; ++vg)
            myT[(vg + 8 * lh) * NP + t * 16 + rowA] = acc[t][vg];
    __syncthreads();

    // ---- epilogue: contract p with per-(e,k) weights; 64 outputs per wave ----
    #pragma unroll
    for (int rep = 0; rep < 2; ++rep) {
        int oidx = lane + rep * 32;          // 0..63
        int m = oidx & 15;                   // channel within tile
        int o = oidx >> 4;                   // output irrep
        const float* Trow = myT + m * NP;
        float v = 0.0f;
        #pragma unroll
        for (int p = 0; p < P1; ++p) v += myW[p * 16 + m]             * Trow[o * P1 + p];
        #pragma unroll
        for (int p = 0; p < P2; ++p) v += myW[(P1 + p) * 16 + m]      * Trow[12 + o * P2 + p];
        #pragma unroll
        for (int p = 0; p < P3; ++p) v += myW[(P1 + P2 + p) * 16 + m] * Trow[56 + o * P3 + p];
        out[b * (D_OUT * MUL) + o * MUL + kBase + m] = v;   // coalesced over lanes
    }
}

// ---------------------------------------------------------------------------
extern "C" void kernel_launch(void* const* d_in, const int* in_sizes, int n_in,
                              void* d_out, int out_size, void* d_ws, size_t ws_size,
                              hipStream_t stream)
{
    const float* x       = (const float*)d_in[0];
    const int*   indices = (const int*)  d_in[1];
    const float* weight  = (const float*)d_in[2];
    const float* U1      = (const float*)d_in[3];
    const float* U2      = (const float*)d_in[4];
    const float* U3      = (const float*)d_in[5];
    float*       out     = (float*)d_out;
    float*       packed  = (float*)d_ws;          // KP*NP floats = 140,800 B

    build_cmat_kernel<<<(KP * NP + 255) / 256, 256, 0, stream>>>(U1, U2, U3, packed);

    const size_t lds_bytes =
        (size_t)(KP * NP + WAVES * ROWS * KP + WAVES * NPAR * 16) * sizeof(float); // ~272 KB < 320 KB/WGP
    const int grid = (NB * MUL) / (WAVES * ROWS);  // 1024 workgroups
    sym_contract_kernel<<<grid, 256, lds_bytes, stream>>>(x, indices, weight, packed, out);
}